// Classifier_8461085573484
// MI455X (gfx1250) — compile-verified
//
#include <hip/hip_runtime.h>
#include <hip/hip_bf16.h>

typedef __attribute__((ext_vector_type(2))) float v2f;
typedef __attribute__((ext_vector_type(8))) float v8f;

#define FDIM 16

// ---------------------------------------------------------------- fill
__global__ void fill_kernel(float* __restrict__ p, float v, long long n) {
    long long i = (long long)blockIdx.x * blockDim.x + threadIdx.x;
    long long stride = (long long)gridDim.x * blockDim.x;
    for (; i < n; i += stride) p[i] = v;
}

// ---------------------------------------------------------------- degree (dst only; self-loop pre-seeded via fill 1.0)
__global__ void degree_kernel(const long long* __restrict__ dst, float* __restrict__ deg,
                              long long nedges) {
    long long e = (long long)blockIdx.x * blockDim.x + threadIdx.x;
    if (e < nedges) atomicAdd(&deg[dst[e]], 1.0f);
}

// ---------------------------------------------------------------- deg -> deg^{-1/2} in place
__global__ void dinv_kernel(float* __restrict__ deg, int n) {
    int i = blockIdx.x * blockDim.x + threadIdx.x;
    if (i < n) {
        float d = deg[i];
        deg[i] = d > 0.0f ? rsqrtf(d) : 0.0f;
    }
}

// ---------------------------------------------------------------- h1 = x (N x 1) * W1 (1 x 16)
__global__ void h1_kernel(const float* __restrict__ x, const float* __restrict__ W1,
                          float* __restrict__ h1, int n) {
    int i = blockIdx.x * blockDim.x + threadIdx.x;
    if (i >= n) return;
    float xv = x[i];
#pragma unroll
    for (int f = 0; f < FDIM; ++f) h1[(long long)i * FDIM + f] = xv * W1[f];
}

// ---------------------------------------------------------------- edge propagation: 16 lanes per edge
// lanes 0..15 of each edge-group read h[s*16 + f] (contiguous 64B) and
// atomic-add into agg[d*16 + f] (contiguous 64B) -> coalesced L2 traffic.
__global__ void edge_prop_kernel(const long long* __restrict__ src,
                                 const long long* __restrict__ dst,
                                 const float* __restrict__ dinv,
                                 const float* __restrict__ h,
                                 float* __restrict__ agg, long long nedges) {
    long long t = (long long)blockIdx.x * blockDim.x + threadIdx.x;
    long long e = t >> 4;
    int f = (int)(t & 15);
    if (e >= nedges) return;
    long long s = src[e];
    long long d = dst[e];
    float w = dinv[s] * dinv[d];
    atomicAdd(&agg[d * FDIM + f], h[s * FDIM + f] * w);
}

// ---------------------------------------------------------------- out = relu(agg + h*dinv^2 + b)  (self-loop msg fused)
__global__ void finalize_relu_kernel(float* __restrict__ agg, const float* __restrict__ h,
                                     const float* __restrict__ dinv, const float* __restrict__ b,
                                     int n) {
    long long t = (long long)blockIdx.x * blockDim.x + threadIdx.x;
    long long i = t >> 4;
    int f = (int)(t & 15);
    if (i >= n) return;
    float di = dinv[i];
    float v = agg[i * FDIM + f] + h[i * FDIM + f] * di * di + b[f];
    agg[i * FDIM + f] = fmaxf(v, 0.0f);
}

// ---------------------------------------------------------------- H2 = A (N x 16) @ W (16 x 16), exact f32 via
// V_WMMA_F32_16X16X4_F32, K=16 chained as 4 WMMAs. One wave per 16-row tile.
// A 16x4 layout (ISA 7.12.2): lanes 0-15: M=lane, v0=K0,v1=K1; lanes 16-31: M=lane-16, v0=K2,v1=K3.
// B 4x16: VGPR v holds row K=v (+2 for hi lanes), N = lane%16.
// C/D:    VGPR v: lanes 0-15 -> M=v, lanes 16-31 -> M=v+8; N = lane%16.
__global__ void gemm_w2_wmma_kernel(const float* __restrict__ A, const float* __restrict__ W,
                                    float* __restrict__ H2, int nrows) {
    int wave = (int)((blockIdx.x * (long long)blockDim.x + threadIdx.x) >> 5);
    int lane = threadIdx.x & 31;
    int r0 = wave * 16;
    if (r0 >= nrows) return;          // wave-uniform: EXEC stays all-ones for WMMA
    int m  = lane & 15;
    int hi = lane >> 4;               // 0: lanes 0-15, 1: lanes 16-31
    v8f c = {};
#pragma unroll
    for (int t = 0; t < 4; ++t) {
        int k0 = 4 * t + 2 * hi;
        v2f a, b;
        a.x = A[(long long)(r0 + m) * FDIM + k0];
        a.y = A[(long long)(r0 + m) * FDIM + k0 + 1];
        b.x = W[k0 * FDIM + m];
        b.y = W[(k0 + 1) * FDIM + m];
        c = __builtin_amdgcn_wmma_f32_16x16x4_f32(
                /*neg_a=*/false, a, /*neg_b=*/false, b,
                /*c_mod=*/(short)0, c, /*reuse_a=*/false, /*reuse_b=*/false);
    }
#pragma unroll
    for (int v = 0; v < 8; ++v) {
        int row = r0 + v + 8 * hi;
        H2[(long long)row * FDIM + m] = c[v];
    }
}

// ---------------------------------------------------------------- finalize layer2 + global mean-pool partial sums
__global__ void finalize_relu_pool_kernel(float* __restrict__ agg, const float* __restrict__ h,
                                          const float* __restrict__ dinv,
                                          const float* __restrict__ b,
                                          float* __restrict__ pooled, int n) {
    __shared__ float sdata[FDIM];
    long long t = (long long)blockIdx.x * blockDim.x + threadIdx.x;
    long long i = t >> 4;
    int f = (int)(t & 15);
    if (threadIdx.x < FDIM) sdata[threadIdx.x] = 0.0f;
    __syncthreads();
    float v = 0.0f;
    if (i < n) {
        float di = dinv[i];
        v = agg[i * FDIM + f] + h[i * FDIM + f] * di * di + b[f];
        v = fmaxf(v, 0.0f);
        agg[i * FDIM + f] = v;
    }
    atomicAdd(&sdata[f], v);          // ds_add_f32
    __syncthreads();
    if (threadIdx.x < FDIM) atomicAdd(&pooled[threadIdx.x], sdata[threadIdx.x]);
}

// ---------------------------------------------------------------- head: mean, linear (16->2), log_softmax
__global__ void head_kernel(const float* __restrict__ pooled, const float* __restrict__ Wl,
                            const float* __restrict__ bl, float* __restrict__ out, float invn) {
    if (threadIdx.x != 0) return;
    float p[FDIM];
#pragma unroll
    for (int f = 0; f < FDIM; ++f) p[f] = pooled[f] * invn;
    float l0 = bl[0], l1 = bl[1];
#pragma unroll
    for (int f = 0; f < FDIM; ++f) {
        l0 += p[f] * Wl[f * 2 + 0];
        l1 += p[f] * Wl[f * 2 + 1];
    }
    float mx  = fmaxf(l0, l1);
    float lse = mx + logf(expf(l0 - mx) + expf(l1 - mx));
    out[0] = l0 - lse;
    out[1] = l1 - lse;
}

// ================================================================ launch
extern "C" void kernel_launch(void* const* d_in, const int* in_sizes, int n_in,
                              void* d_out, int out_size, void* d_ws, size_t ws_size,
                              hipStream_t stream) {
    const float*     x   = (const float*)d_in[0];
    const long long* ei  = (const long long*)d_in[1];   // int64 edge_index [2, E]
    const float*     W1  = (const float*)d_in[2];
    const float*     b1  = (const float*)d_in[3];
    const float*     W2  = (const float*)d_in[4];
    const float*     b2  = (const float*)d_in[5];
    const float*     Wl  = (const float*)d_in[6];
    const float*     bl  = (const float*)d_in[7];
    float*           out = (float*)d_out;

    const int       n = in_sizes[0];                 // x is [n, 1]
    const long long E = (long long)in_sizes[1] / 2;
    const long long* src = ei;
    const long long* dst = ei + E;

    // workspace layout (floats)
    float* ws     = (float*)d_ws;
    float* dinv   = ws;                              // [n]      degree -> deg^{-1/2}
    float* h1     = dinv + n;                        // [n*16]
    float* agg1   = h1 + (long long)n * FDIM;        // [n*16]   becomes out1 in place
    float* h2     = agg1 + (long long)n * FDIM;      // [n*16]
    float* agg2   = h2 + (long long)n * FDIM;        // [n*16]   becomes out2 in place
    float* pooled = agg2 + (long long)n * FDIM;      // [16]

    const int B = 256;
    const long long nf  = (long long)n * FDIM;
    const unsigned gN   = (unsigned)((n + B - 1) / B);
    const unsigned gNF  = (unsigned)((nf + B - 1) / B);
    const unsigned gE   = (unsigned)((E + B - 1) / B);
    const unsigned gE16 = (unsigned)((E * FDIM + B - 1) / B);

    // init accumulators (deg seeded with 1.0 for the self loop)
    fill_kernel<<<gN, B, 0, stream>>>(dinv, 1.0f, n);
    fill_kernel<<<gNF, B, 0, stream>>>(agg1, 0.0f, nf);
    fill_kernel<<<gNF, B, 0, stream>>>(agg2, 0.0f, nf);
    fill_kernel<<<1, FDIM, 0, stream>>>(pooled, 0.0f, FDIM);

    // normalization
    degree_kernel<<<gE, B, 0, stream>>>(dst, dinv, E);
    dinv_kernel<<<gN, B, 0, stream>>>(dinv, n);

    // layer 1
    h1_kernel<<<gN, B, 0, stream>>>(x, W1, h1, n);
    edge_prop_kernel<<<gE16, B, 0, stream>>>(src, dst, dinv, h1, agg1, E);
    finalize_relu_kernel<<<gNF, B, 0, stream>>>(agg1, h1, dinv, b1, n);

    // layer 2: h2 = out1 @ W2 on the matrix pipe (f32 WMMA), then propagate
    {
        int waves = (n + 15) / 16;
        long long threads = (long long)waves * 32;
        unsigned gW = (unsigned)((threads + B - 1) / B);
        gemm_w2_wmma_kernel<<<gW, B, 0, stream>>>(agg1, W2, h2, n);
    }
    edge_prop_kernel<<<gE16, B, 0, stream>>>(src, dst, dinv, h2, agg2, E);
    finalize_relu_pool_kernel<<<gNF, B, 0, stream>>>(agg2, h2, dinv, b2, pooled, n);

    // head
    head_kernel<<<1, 32, 0, stream>>>(pooled, Wl, bl, out, 1.0f / (float)n);
}